// FCM_series_1_17892833755314
// MI455X (gfx1250) — compile-verified
//
#include <hip/hip_runtime.h>
#include <math.h>

// ---------------------------------------------------------------------------
// Problem constants (from reference): LAG=13 -> 12 lags, B=32, N=1200, H=1/3
// Flattened GEMM: M = 12*32 = 384 rows, K = 1200, N = 1200 cols.
// Tiling: block = 256 threads = 8 waves; wave = 16 rows x 48 cols (3 N-tiles)
//         block = 128 rows x 48 cols;  grid = (1200/48, 384/128) = (25, 3)
// ---------------------------------------------------------------------------
#define NN   1200
#define MM   384          // 12 lags * 32 batch
#define NLAG 12
#define NB   32
#define JT   48           // N-columns per block (3 x 16 WMMA tiles)

typedef __attribute__((ext_vector_type(2))) float v2f;
typedef __attribute__((ext_vector_type(8))) float v8f;

// gamma_fn from the reference: exp(lgamma(x)) for x>0, reflection otherwise.
__device__ __forceinline__ float gamma_dev(float x) {
    if (x > 0.0f) {
        return expf(lgammaf(x));
    } else {
        const float pi = 3.14159265358979323846f;
        return pi / (sinf(pi * x) * expf(lgammaf(1.0f - x)));
    }
}

// ---------------------------------------------------------------------------
// Pack train_init[..., l, 1] and [..., l+1, 1] into row-contiguous (m, i)
// layouts so WMMA A-fragments are contiguous b64 loads.
// ---------------------------------------------------------------------------
__global__ void fcm_pack(const float* __restrict__ ti,
                         float* __restrict__ X1p,
                         float* __restrict__ X2p) {
    int idx = blockIdx.x * blockDim.x + threadIdx.x;
    if (idx >= MM * NN) return;
    int m = idx / NN;
    int i = idx - m * NN;
    int l = m >> 5;        // /32
    int b = m & 31;
    size_t base = (size_t)(b * NN + i) * 13;
    X1p[idx] = ti[(base + l) * 2 + 1];
    X2p[idx] = ti[(base + l + 1) * 2 + 1];
}

// ---------------------------------------------------------------------------
// Per-lag scalars: belta[l], c2[l] = g_a1/(Gamma(4)*Gamma(alpha-2)),
// invHf[l] = 1 / H^fract[l].
// ---------------------------------------------------------------------------
__global__ void fcm_scalars(const float* __restrict__ alpha,
                            const float* __restrict__ fract,
                            float* __restrict__ sc) {
    int l = threadIdx.x;
    if (l >= NLAG) return;
    float al = alpha[l];
    float fr = fract[l];
    float ga1 = gamma_dev(al + 1.0f);
    float belta = 0.0f;
#pragma unroll
    for (int kk = 0; kk < 4; ++kk) {
        belta += ga1 / (gamma_dev((float)kk + 1.0f) * gamma_dev(al - (float)kk + 1.0f));
    }
    float c2    = ga1 / (gamma_dev(4.0f) * gamma_dev(al - 2.0f));
    float invHf = 1.0f / powf(1.0f / 3.0f, fr);
    sc[l * 3 + 0] = belta;
    sc[l * 3 + 1] = c2;
    sc[l * 3 + 2] = invHf;
}

// ---------------------------------------------------------------------------
// Fused triple-GEMM + epilogue with V_WMMA_F32_16X16X4_F32.
//
// Operand layout (ISA 7.12.2):
//   A 16x4 : lane r / r+16 holds A[M=r][K = 2*half + {0,1}] in the 2 VGPRs
//   B 4x16 : lane r / r+16 holds B[K = 2*half + {0,1}][N=r]
//   C 16x16: VGPR v holds C[M = v + 8*half][N = r]
//
// Each wave: one 16-row A panel (x3 matrices), three 16-col B tiles from the
// LDS-staged WW panel -> 9 WMMAs per 3 global b64 loads per k-substep.
// WW panel (16 K x 48 N) is double-buffered through registers so the next
// chunk's global loads overlap the current chunk's 36 WMMAs.
// ---------------------------------------------------------------------------
__global__ __launch_bounds__(256)
void fcm_main(const float* __restrict__ Ap,   // (384,1200) aggregate input
              const float* __restrict__ Wp,   // WW (1200,1200)
              const float* __restrict__ X2p,  // -> A_0_NEW
              const float* __restrict__ X1p,  // -> A_N_OLD
              const float* __restrict__ sc,   // (12,3) belta, c2, invHf
              const float* __restrict__ lam,  // lambd (12,200)
              const float* __restrict__ lco,  // l     (12,200)
              float* __restrict__ out) {      // (32,12,1200)
    __shared__ float ldsB[16][JT];

    const int tid  = threadIdx.x;
    const int wave = tid >> 5;
    const int lane = tid & 31;
    const int half = lane >> 4;
    const int r    = lane & 15;

    const int jBase = blockIdx.x * JT;
    const int mBase = blockIdx.y * 128 + wave * 16;
    const int mRow  = mBase + r;

    const float* a0 = Ap  + (size_t)mRow * NN;
    const float* a1 = X2p + (size_t)mRow * NN;
    const float* a2 = X1p + (size_t)mRow * NN;

    v8f acc[3][3];                      // [matrix][n-tile]
#pragma unroll
    for (int m = 0; m < 3; ++m)
#pragma unroll
        for (int n = 0; n < 3; ++n)
            acc[m][n] = (v8f){};

    // WW-panel loader assignment: 768 elements, 3 per thread.
    int ldr[3], ldc[3];
#pragma unroll
    for (int t = 0; t < 3; ++t) {
        int idx = tid + t * 256;
        ldr[t] = idx / JT;
        ldc[t] = idx - ldr[t] * JT;
    }

    // Preload chunk 0 into registers.
    float cur[3];
#pragma unroll
    for (int t = 0; t < 3; ++t)
        cur[t] = Wp[(size_t)ldr[t] * NN + jBase + ldc[t]];

    for (int k = 0; k < NN; k += 16) {
        __syncthreads();
#pragma unroll
        for (int t = 0; t < 3; ++t)
            ldsB[ldr[t]][ldc[t]] = cur[t];
        __syncthreads();

        // Fetch next chunk while this chunk's WMMAs run.
        if (k + 16 < NN) {
#pragma unroll
            for (int t = 0; t < 3; ++t)
                cur[t] = Wp[(size_t)(k + 16 + ldr[t]) * NN + jBase + ldc[t]];
#pragma unroll
            for (int t = 0; t < 3; ++t)
                if (k + 32 < NN)
                    __builtin_prefetch(&Wp[(size_t)(k + 32 + ldr[t]) * NN + jBase + ldc[t]], 0, 3);
        }

#pragma unroll
        for (int s = 0; s < 4; ++s) {
            const int ko = k + 4 * s + 2 * half;
            const int br = 4 * s + 2 * half;
            v2f af = *(const v2f*)(a0 + ko);
            v2f xf = *(const v2f*)(a1 + ko);
            v2f yf = *(const v2f*)(a2 + ko);
#pragma unroll
            for (int n = 0; n < 3; ++n) {
                v2f b;
                b.x = ldsB[br][n * 16 + r];
                b.y = ldsB[br + 1][n * 16 + r];
                acc[0][n] = __builtin_amdgcn_wmma_f32_16x16x4_f32(
                    false, af, false, b, (short)0, acc[0][n], false, false);
                acc[1][n] = __builtin_amdgcn_wmma_f32_16x16x4_f32(
                    false, xf, false, b, (short)0, acc[1][n], false, false);
                acc[2][n] = __builtin_amdgcn_wmma_f32_16x16x4_f32(
                    false, yf, false, b, (short)0, acc[2][n], false, false);
            }
        }
    }

    // Epilogue: fused tanh + fractional-calculus scaling, direct store.
#pragma unroll
    for (int v = 0; v < 8; ++v) {
        const int m    = mBase + v + 8 * half;
        const int lidx = m >> 5;   // lag
        const int bidx = m & 31;   // batch
        const float belta = sc[lidx * 3 + 0];
        const float cc2   = sc[lidx * 3 + 1];
        const float invHf = sc[lidx * 3 + 2];
#pragma unroll
        for (int n = 0; n < 3; ++n) {
            const int j  = jBase + n * 16 + r;
            const int jm = j % 200;
            const float t0 = tanhf(-acc[0][n][v]);
            const float t1 = tanhf(-acc[1][n][v]);
            const float t2 = tanhf(-acc[2][n][v]);
            const float lamv = lam[lidx * 200 + jm];
            const float lv   = lco[lidx * 200 + jm];
            // res = lambd/belta * (2*out/H^fract + l*(A_0_NEW + c2*A_N_OLD)/H)
            const float res = lamv / belta *
                              (2.0f * t0 * invHf + lv * (t1 + cc2 * t2) * 3.0f);
            out[((size_t)bidx * NLAG + lidx) * NN + j] = res;
        }
    }
}

// ---------------------------------------------------------------------------
// Launch
// ---------------------------------------------------------------------------
extern "C" void kernel_launch(void* const* d_in, const int* in_sizes, int n_in,
                              void* d_out, int out_size, void* d_ws, size_t ws_size,
                              hipStream_t stream) {
    const float* A     = (const float*)d_in[0];  // (12,32,1200,1)
    const float* WW    = (const float*)d_in[1];  // (1200,1200,1)
    const float* ti    = (const float*)d_in[2];  // (32,1200,13,2)
    const float* alpha = (const float*)d_in[3];  // (12,1)
    const float* fract = (const float*)d_in[4];  // (12,1)
    const float* lambd = (const float*)d_in[5];  // (12,1,200,1)
    const float* lco   = (const float*)d_in[6];  // (12,1,200,1)
    // d_in[7] (A_y_list zeros) is unused: index NUM=0 slot is recomputed.
    float* out = (float*)d_out;

    float* X1p = (float*)d_ws;          // 460800 f32
    float* X2p = X1p + MM * NN;         // 460800 f32
    float* sc  = X2p + MM * NN;         // 36 f32

    fcm_pack<<<(MM * NN + 255) / 256, 256, 0, stream>>>(ti, X1p, X2p);
    fcm_scalars<<<1, 32, 0, stream>>>(alpha, fract, sc);

    dim3 grid(NN / JT, MM / 128);       // 25 x 3
    fcm_main<<<grid, 256, 0, stream>>>(A, WW, X2p, X1p, sc, lambd, lco, out);
}